// DeepseekAllReduce_56994216018564
// MI455X (gfx1250) — compile-verified
//
#include <hip/hip_runtime.h>

// ---------------------------------------------------------------------------
// Fused TP all-reduce (+4 partials) + residual add + RMSNorm, fp32.
// Bandwidth-bound (~821 MB @ 23.3 TB/s). Uses CDNA5 async global->LDS data
// mover (ASYNCcnt) as a VGPR-free deep load queue; wave-private LDS slices so
// the streaming phase needs zero barriers. Single-use streams are tagged
// non-temporal (cpol TH_NT=1) so the 192 MB L2 stays hot for norm_weight.
// ---------------------------------------------------------------------------

#define TP       4
#define HIDDEN   7168
#define EPS_F    1e-6f
#define BLOCK    256
#define VEC      4
#define CHUNK    (BLOCK * VEC)      // 1024 floats per chunk
#define NCHUNK   (HIDDEN / CHUNK)   // 7 chunks per row
#define NSTREAM  (TP + 1)           // 4 rank partials + residual
#define NBUF     3                  // triple-buffered pipeline
#define CPOL_NT  1                  // gfx12+ CPol: TH[2:0]=1 -> non-temporal

typedef float v4f __attribute__((ext_vector_type(4)));
typedef int   v4i __attribute__((vector_size(4 * sizeof(int))));

// Address-space-qualified pointer types matching the async-LDS builtin:
//   param0: v4i in global (AS1), param1: v4i in LDS (AS3)
typedef __attribute__((address_space(1))) v4i gas_v4i;
typedef __attribute__((address_space(3))) v4i las_v4i;

#if defined(__has_builtin)
#if __has_builtin(__builtin_amdgcn_global_load_async_to_lds_b128) && \
    __has_builtin(__builtin_amdgcn_s_wait_asynccnt)
#define USE_ASYNC 1
#endif
#endif
#ifndef USE_ASYNC
#define USE_ASYNC 0
#endif

__global__ __launch_bounds__(BLOCK) void fused_ar_rmsnorm_kernel(
    const float* __restrict__ hs,        // [TP, tokens, HIDDEN]
    const float* __restrict__ resid,     // [tokens, HIDDEN]
    const float* __restrict__ w,         // [HIDDEN]
    float* __restrict__ normed,          // [tokens, HIDDEN]
    float* __restrict__ resid_out,       // [tokens, HIDDEN]
    int tokens)
{
    const int tid = threadIdx.x;
    const size_t row         = (size_t)blockIdx.x * HIDDEN;
    const size_t rank_stride = (size_t)tokens * HIDDEN;

    const float* src[NSTREAM];
    src[0] = hs + row;
    src[1] = src[0] + rank_stride;
    src[2] = src[1] + rank_stride;
    src[3] = src[2] + rank_stride;
    src[4] = resid + row;

    __shared__ float red[BLOCK / 32];

    v4f   acc[NCHUNK];
    float ss = 0.0f;

#if USE_ASYNC
    // [buf][stream][CHUNK] staging; each lane only ever touches its own 16B
    // per stream, so waves are fully independent in the streaming phase.
    __shared__ float lds[NBUF][NSTREAM][CHUNK];

    auto issue_chunk = [&](int c, int buf) {
        const int e = c * CHUNK + tid * VEC;   // element offset within row
        const int l = tid * VEC;               // element offset within buffer
#pragma unroll
        for (int s = 0; s < NSTREAM; ++s) {
            __builtin_amdgcn_global_load_async_to_lds_b128(
                (gas_v4i*)(src[s] + e),
                (las_v4i*)&lds[buf][s][l],
                0, CPOL_NT);
        }
    };

    issue_chunk(0, 0);
    issue_chunk(1, 1);

#pragma unroll
    for (int c = 0; c < NCHUNK; ++c) {
        if (c + 2 < NCHUNK) {
            issue_chunk(c + 2, (c + 2) % NBUF);
            // Async loads complete in order: <= 2*NSTREAM outstanding
            // guarantees chunk c's NSTREAM loads have landed in LDS.
            __builtin_amdgcn_s_wait_asynccnt(2 * NSTREAM);
        } else if (c + 1 < NCHUNK) {
            __builtin_amdgcn_s_wait_asynccnt(NSTREAM);
        } else {
            __builtin_amdgcn_s_wait_asynccnt(0);
        }

        const int buf = c % NBUF;
        const int l   = tid * VEC;
        v4f a0 = *(const v4f*)&lds[buf][0][l];
        v4f a1 = *(const v4f*)&lds[buf][1][l];
        v4f a2 = *(const v4f*)&lds[buf][2][l];
        v4f a3 = *(const v4f*)&lds[buf][3][l];
        v4f r  = *(const v4f*)&lds[buf][4][l];

        v4f s = a0 + a1 + a2 + a3 + r;
        acc[c] = s;
        ss += s.x * s.x + s.y * s.y + s.z * s.z + s.w * s.w;

        // residual_out is never re-read by this kernel: stream it out NT.
        __builtin_nontemporal_store(
            s, (v4f*)(resid_out + row + (size_t)c * CHUNK + tid * VEC));
    }
#else
    // Fallback: direct vectorized non-temporal loads.
#pragma unroll
    for (int c = 0; c < NCHUNK; ++c) {
        const int e = c * CHUNK + tid * VEC;
        v4f a0 = __builtin_nontemporal_load((const v4f*)(src[0] + e));
        v4f a1 = __builtin_nontemporal_load((const v4f*)(src[1] + e));
        v4f a2 = __builtin_nontemporal_load((const v4f*)(src[2] + e));
        v4f a3 = __builtin_nontemporal_load((const v4f*)(src[3] + e));
        v4f r  = __builtin_nontemporal_load((const v4f*)(src[4] + e));

        v4f s = a0 + a1 + a2 + a3 + r;
        acc[c] = s;
        ss += s.x * s.x + s.y * s.y + s.z * s.z + s.w * s.w;

        __builtin_nontemporal_store(s, (v4f*)(resid_out + row + e));
    }
#endif

    // ---- block-wide sum-of-squares reduction (wave32) -----------------------
#pragma unroll
    for (int off = 16; off > 0; off >>= 1)
        ss += __shfl_xor(ss, off, 32);
    if ((tid & 31) == 0)
        red[tid >> 5] = ss;
    __syncthreads();

    float tot = 0.0f;
#pragma unroll
    for (int i = 0; i < BLOCK / 32; ++i)
        tot += red[i];

    const float scale = rsqrtf(tot * (1.0f / (float)HIDDEN) + EPS_F);

    // ---- scale by weight, write normed (NT; never re-read) ------------------
#pragma unroll
    for (int c = 0; c < NCHUNK; ++c) {
        const int e = c * CHUNK + tid * VEC;
        v4f wv = *(const v4f*)(w + e);   // hot in L2 across all blocks
        v4f o  = acc[c] * scale * wv;
        __builtin_nontemporal_store(o, (v4f*)(normed + row + e));
    }
}

extern "C" void kernel_launch(void* const* d_in, const int* in_sizes, int n_in,
                              void* d_out, int out_size, void* d_ws, size_t ws_size,
                              hipStream_t stream) {
    const float* hs    = (const float*)d_in[0];  // [TP, tokens, HIDDEN]
    const float* resid = (const float*)d_in[1];  // [tokens, HIDDEN]
    const float* w     = (const float*)d_in[2];  // [HIDDEN]

    const int tokens = in_sizes[1] / HIDDEN;

    float* normed    = (float*)d_out;                            // output 0
    float* resid_out = (float*)d_out + (size_t)tokens * HIDDEN;  // output 1

    fused_ar_rmsnorm_kernel<<<tokens, BLOCK, 0, stream>>>(
        hs, resid, w, normed, resid_out, tokens);
}